// Pool_layer_27178553049382
// MI455X (gfx1250) — compile-verified
//
#include <hip/hip_runtime.h>

typedef __attribute__((ext_vector_type(2))) float v2f;
typedef __attribute__((ext_vector_type(8))) float v8f;

#define BS      16
#define NPTS    2048
#define NCH     256
#define POOLN   512
#define K_NBR   16
#define TOPK    17                       // 16 neighbors + self
#define CHUNK   256                      // candidates staged per chunk
#define NCHUNK  (NPTS / CHUNK)           // 8
#define DSTRIDE 272                      // padded dist row stride (16 mod 64 -> conflict-free scan)
#define NTHREADS 256

// shared float pool layout (floats):
//   [0,2048)      quad            --+
//   [2048,4096)   cx                | reused after main loop as:
//   [4096,6144)   cy                |   mergeD [0,4352)  (16*16 scanners * 17)
//   [6144,8192)   cz                |   mergeI [4352,8704)
//   [8192,12544)  dist 16 x 272   --+ (tail of mergeI overlaps dist head; barrier-ordered)
#define SMEM_FLOATS (8192 + 16 * DSTRIDE)

__global__ __launch_bounds__(NTHREADS) void knn_pool_kernel(
    const float* __restrict__ vertices,     // [BS][NPTS][3]
    const float* __restrict__ feature_map,  // [BS][NPTS][NCH]
    const int*   __restrict__ sample_idx,   // [POOLN]
    float* __restrict__ out_v,              // [BS][POOLN][3]
    float* __restrict__ out_f)              // [BS][POOLN][NCH]
{
    __shared__ float smem[SMEM_FLOATS];
    __shared__ float qx[16], qy[16], qz[16], qquadL[16];
    __shared__ int   qsidx[16];
    __shared__ int   nbrLds[16][K_NBR];

    float* quadL  = smem;
    float* cxL    = smem + 2048;
    float* cyL    = smem + 4096;
    float* czL    = smem + 6144;
    float* distL  = smem + 8192;
    float* mergeD = smem;                  // aliases quad/cx (dead after main loop)
    int*   mergeI = (int*)(smem + 4352);   // aliases cy/cz/dist head

    const int tid  = threadIdx.x;
    const int lane = tid & 31;
    const int wave = tid >> 5;
    const int qt   = blockIdx.x;           // query tile 0..31
    const int b    = blockIdx.y;           // batch 0..15
    const int qi   = tid >> 4;             // query handled in scan/gather phases
    const int sub  = tid & 15;             // scanner / channel-group id

    const float* vb = vertices + (size_t)b * NPTS * 3;

    // ---- stage SoA coords + squared norms in LDS (one pass over 24 KB) ----
    for (int i = tid; i < NPTS; i += NTHREADS) {
        float x = vb[i * 3 + 0];
        float y = vb[i * 3 + 1];
        float z = vb[i * 3 + 2];
        cxL[i] = x; cyL[i] = y; czL[i] = z;
        quadL[i] = x * x + y * y + z * z;
    }
    __syncthreads();

    // ---- query data straight from LDS ----
    if (tid < 16) {
        int s = sample_idx[qt * 16 + tid];
        qsidx[tid]  = s;
        qx[tid]     = cxL[s];
        qy[tid]     = cyL[s];
        qz[tid]     = czL[s];
        qquadL[tid] = quadL[s];
    }
    __syncthreads();

    // ---- vertices_pool output (trivial slice) ----
    if (tid < 48) {
        int qi2 = tid / 3, cc = tid % 3;
        out_v[((size_t)b * POOLN + qt * 16 + qi2) * 3 + cc] =
            vb[(size_t)qsidx[qi2] * 3 + cc];
    }

    // ---- WMMA A operand (16x4 f32, K=3 zero-padded), loop-invariant ----
    // lanes 0-15: M=lane {K0,K1}; lanes 16-31: M=lane-16 {K2,K3}
    v2f aOp;
    {
        const int m = lane & 15;
        aOp.x = (lane < 16) ? qx[m] : qz[m];
        aOp.y = (lane < 16) ? qy[m] : 0.0f;
    }
    // per-lane query norms for the distance epilogue, hoisted to registers
    float qq[8];
#pragma unroll
    for (int r = 0; r < 8; ++r) qq[r] = qquadL[r + ((lane < 16) ? 0 : 8)];

    // ---- private sorted top-17 per scanner thread (128 candidates each) ----
    float best_d[TOPK];
    int   best_i[TOPK];
#pragma unroll
    for (int s = 0; s < TOPK; ++s) { best_d[s] = __builtin_inff(); best_i[s] = 0x7fffffff; }

#pragma unroll 1
    for (int ch = 0; ch < NCHUNK; ++ch) {
        const int chunkBase = ch * CHUNK;
#pragma unroll
        for (int t = 0; t < 2; ++t) {
            const int tile = wave * 2 + t;          // 0..15 within chunk
            const int n0   = chunkBase + tile * 16;
            const int n    = lane & 15;
            const int cand = n0 + n;

            // B operand from LDS: address-select, no exec divergence
            const float* bxPtr = (lane < 16) ? (cxL + cand) : (czL + cand);
            v2f bOp;
            bOp.x = *bxPtr;
            bOp.y = (lane < 16) ? cyL[cand] : 0.0f;

            v8f acc = {};
            acc = __builtin_amdgcn_wmma_f32_16x16x4_f32(
                /*neg_a=*/false, aOp, /*neg_b=*/false, bOp,
                /*c_mod=*/(short)0, acc, /*reuse_a=*/false, /*reuse_b=*/false);

            // dist = ||q||^2 + ||c||^2 - 2*inner; C/D layout:
            // vgpr r, lane L -> M = r + (L<16?0:8), N = L&15
            const float cq = quadL[cand];
#pragma unroll
            for (int r = 0; r < 8; ++r) {
                const int m = r + ((lane < 16) ? 0 : 8);
                distL[m * DSTRIDE + tile * 16 + n] = qq[r] + cq - 2.0f * acc[r];
            }
        }
        __syncthreads();

        // parallel selection: 16 scanners per query; single compact insert body
        {
            const float* drow = distL + qi * DSTRIDE;
#pragma unroll 1
            for (int jj = 0; jj < 16; ++jj) {
                const int   j  = sub + jj * 16;
                const float dv = drow[j];
                if (dv < best_d[TOPK - 1]) {           // rare-path guard
                    float cd = dv;
                    int   ci = chunkBase + j;
#pragma unroll
                    for (int s = 0; s < TOPK; ++s) {   // branchless sorted insert
                        const bool  lt   = cd < best_d[s];
                        const float dmin = lt ? cd : best_d[s];
                        const float dmax = lt ? best_d[s] : cd;
                        const int   imin = lt ? ci : best_i[s];
                        const int   imax = lt ? best_i[s] : ci;
                        best_d[s] = dmin; best_i[s] = imin;
                        cd = dmax; ci = imax;
                    }
                }
            }
        }
        __syncthreads();
    }

    // ---- publish per-scanner partials (aliased scratch; barrier-ordered) ----
    {
        const int base = tid * TOPK;
#pragma unroll
        for (int s = 0; s < TOPK; ++s) {
            mergeD[base + s] = best_d[s];
            mergeI[base + s] = best_i[s];
        }
    }
    __syncthreads();

    // ---- lexicographic (d, idx) merge: exact jax.lax.top_k semantics ----
    if (tid < 16) {
        float fd[TOPK];
        int   fi[TOPK];
#pragma unroll
        for (int s = 0; s < TOPK; ++s) { fd[s] = __builtin_inff(); fi[s] = 0x7fffffff; }
        const int mb = tid * 16 * TOPK;
#pragma unroll 1
        for (int e = 0; e < 16 * TOPK; ++e) {
            float cd = mergeD[mb + e];
            int   ci = mergeI[mb + e];
            if (cd < fd[TOPK - 1] || (cd == fd[TOPK - 1] && ci < fi[TOPK - 1])) {
#pragma unroll
                for (int s = 0; s < TOPK; ++s) {       // branchless lex insert
                    const bool  lt   = (cd < fd[s]) || (cd == fd[s] && ci < fi[s]);
                    const float dmin = lt ? cd : fd[s];
                    const float dmax = lt ? fd[s] : cd;
                    const int   imin = lt ? ci : fi[s];
                    const int   imax = lt ? fi[s] : ci;
                    fd[s] = dmin; fi[s] = imin;
                    cd = dmax; ci = imax;
                }
            }
        }
#pragma unroll
        for (int k = 0; k < K_NBR; ++k) nbrLds[tid][k] = fi[k + 1];  // drop self
    }
    __syncthreads();

    // ---- gather + max pooling: thread = (query, 16-channel group) ----
    const float* fb = feature_map + (size_t)b * NPTS * NCH;

    float4 a0, a1, a2, a3;
    a0 = a1 = a2 = a3 = make_float4(-__builtin_inff(), -__builtin_inff(),
                                    -__builtin_inff(), -__builtin_inff());
#pragma unroll
    for (int k = 0; k < K_NBR; ++k) {
        const int nb = nbrLds[qi][k];
        const float4* row = (const float4*)(fb + (size_t)nb * NCH + sub * 16);
        float4 f0 = row[0], f1 = row[1], f2 = row[2], f3 = row[3];
        a0.x = fmaxf(a0.x, f0.x); a0.y = fmaxf(a0.y, f0.y); a0.z = fmaxf(a0.z, f0.z); a0.w = fmaxf(a0.w, f0.w);
        a1.x = fmaxf(a1.x, f1.x); a1.y = fmaxf(a1.y, f1.y); a1.z = fmaxf(a1.z, f1.z); a1.w = fmaxf(a1.w, f1.w);
        a2.x = fmaxf(a2.x, f2.x); a2.y = fmaxf(a2.y, f2.y); a2.z = fmaxf(a2.z, f2.z); a2.w = fmaxf(a2.w, f2.w);
        a3.x = fmaxf(a3.x, f3.x); a3.y = fmaxf(a3.y, f3.y); a3.z = fmaxf(a3.z, f3.z); a3.w = fmaxf(a3.w, f3.w);
    }
    const int qrow = qt * 16 + qi;
    float4* orow = (float4*)(out_f + ((size_t)b * POOLN + qrow) * NCH + sub * 16);
    orow[0] = a0; orow[1] = a1; orow[2] = a2; orow[3] = a3;
}

extern "C" void kernel_launch(void* const* d_in, const int* in_sizes, int n_in,
                              void* d_out, int out_size, void* d_ws, size_t ws_size,
                              hipStream_t stream) {
    const float* vertices    = (const float*)d_in[0];
    const float* feature_map = (const float*)d_in[1];
    const int*   sample_idx  = (const int*)d_in[2];

    float* out   = (float*)d_out;
    float* out_v = out;                                   // [16][512][3]
    float* out_f = out + (size_t)BS * POOLN * 3;          // [16][512][256]

    dim3 grid(POOLN / 16, BS);   // 32 query tiles x 16 batches
    knn_pool_kernel<<<grid, NTHREADS, 0, stream>>>(vertices, feature_map, sample_idx,
                                                   out_v, out_f);
}